// DETM_85701777424992
// MI455X (gfx1250) — compile-verified
//
#include <hip/hip_runtime.h>
#include <hip/hip_bf16.h>

// ---------------- problem constants (match reference) ----------------
#define KK   50      // topics
#define TT   40      // time steps
#define VV   30000   // vocab
#define THH  800     // theta hidden
#define EHH  200     // eta hidden
#define EE   300     // embedding
#define LL   3       // lstm layers
#define BB   100     // batch
#define DELTA_C 0.005f
#define EPS_C   1e-6f

#define KPAD 30052           // (VV + KK) padded to multiple of 4
#define VT   128             // v-columns per logits chunk
#define NCH  ((VV + VT - 1) / VT)   // 235

typedef float v2f __attribute__((ext_vector_type(2)));
typedef float v8f __attribute__((ext_vector_type(8)));

__device__ __forceinline__ float sigf(float x) { return 1.0f / (1.0f + expf(-x)); }

// ---------------- workspace layout (floats) ----------------
#define OFF_OUT0  ((size_t)0)                       // T*EH          = 8000
#define OFF_X0    (OFF_OUT0 + (size_t)TT*EHH)       // 8000
#define OFF_X1    (OFF_X0   + (size_t)TT*EHH)       // 8000
#define OFF_ETAS  (OFF_X1   + (size_t)TT*EHH)       // T*K = 2000
#define OFF_KLETA (OFF_ETAS + (size_t)TT*KK)        // 1
#define OFF_INP   (OFF_KLETA + 4)                   // B*KPAD = 3,005,200 (even offset!)
#define OFF_H     (OFF_INP  + (size_t)BB*KPAD)      // B*TH = 80,000
#define OFF_THETA (OFF_H    + (size_t)BB*THH)       // B*K = 5000
#define OFF_KLTH  (OFF_THETA + (size_t)BB*KK)       // B = 100
#define OFF_ZPART (OFF_KLTH + BB)                   // T*NCH*2*64 = 1,203,200
#define OFF_Z     (OFF_ZPART + (size_t)TT*NCH*2*64) // T*K = 2000
#define OFF_NLLP  (OFF_Z    + (size_t)TT*KK)        // T*NCH = 9400
#define OFF_KLAP  (OFF_NLLP + (size_t)TT*NCH)       // 2344
#define KLA_NEL   (KK*TT*EE)                        // 600,000
#define KLA_BLK   ((KLA_NEL + 255)/256)             // 2344
#define OFF_S     (OFF_KLAP + KLA_BLK)              // 4 scalars

// =====================================================================
// Generic WMMA fp32 GEMM:  C[M,N] = act( A[M,Kd] * B[N,Kd]^T + bias[N] )
// Block = 256 threads = 8 waves; wave (mt=w>>2, nt=w&3) -> 16x16 tile;
// block tile = 32(M) x 64(N).
//
// Branchless hot loop: out-of-range rows/cols are CLAMPED to valid
// addresses (their outputs are discarded at the store guard — D row m
// depends only on A row m, D col n only on B row n). Only the inner-dim
// tail (ka >= KdReal) needs true zeroing, peeled outside the loop.
// All strides/offsets are even -> 8B-aligned float2 loads.
// =====================================================================
__global__ __launch_bounds__(256)
void k_gemm_wmma(const float* __restrict__ A, int lda,
                 const float* __restrict__ Bm, int ldb,
                 const float* __restrict__ bias,
                 float* __restrict__ C, int ldc,
                 int M, int N, int KdReal, int act)
{
    const int lane = threadIdx.x & 31;
    const int wave = threadIdx.x >> 5;
    const int half = lane >> 4;
    const int l16  = lane & 15;
    const int mt = wave >> 2;            // 0..1
    const int nt = wave & 3;             // 0..3
    const int m0 = blockIdx.x * 32 + mt * 16;
    const int n0 = blockIdx.y * 64 + nt * 16;
    const int am = m0 + l16;             // A row for this lane
    const int bn = n0 + l16;             // B row (C column) for this lane
    // clamped, always-valid base pointers (+2*half = lane's K offset)
    const float* __restrict__ Ap =
        A + (size_t)(am < M ? am : M - 1) * (size_t)lda + 2 * half;
    const float* __restrict__ Bp =
        Bm + (size_t)(bn < N ? bn : N - 1) * (size_t)ldb + 2 * half;

    const int KdMain = KdReal & ~3;

    v8f acc = {0.f,0.f,0.f,0.f,0.f,0.f,0.f,0.f};
#pragma unroll 2
    for (int kk = 0; kk < KdMain; kk += 4) {
        const v2f av = *(const v2f*)(Ap + kk);
        const v2f bv = *(const v2f*)(Bp + kk);
        acc = __builtin_amdgcn_wmma_f32_16x16x4_f32(
            false, av, false, bv, (short)0, acc, false, false);
    }
    if (KdMain < KdReal) {               // tail step: true zeroing required
        const int ka = KdMain + 2 * half;
        float a0 = 0.f, a1 = 0.f, b0 = 0.f, b1 = 0.f;
        if (ka     < KdReal) { a0 = Ap[KdMain];     b0 = Bp[KdMain];     }
        if (ka + 1 < KdReal) { a1 = Ap[KdMain + 1]; b1 = Bp[KdMain + 1]; }
        v2f av = {a0, a1};
        v2f bv = {b0, b1};
        acc = __builtin_amdgcn_wmma_f32_16x16x4_f32(
            false, av, false, bv, (short)0, acc, false, false);
    }
#pragma unroll
    for (int i = 0; i < 8; ++i) {
        const int row = m0 + 8 * half + i;
        const int col = n0 + l16;
        if (row < M && col < N) {
            float v = acc[i] + (bias ? bias[col] : 0.f);
            if (act == 1) v = tanhf(v);
            C[(size_t)row * ldc + col] = v;
        }
    }
}

// =====================================================================
// logits tile helper: 64(k-rows) x 128(v-cols) per block at (t, chunk).
// Wave w: mt = w>>1 (0..3), ng = w&1; wave owns 4 n-tiles (ng*4+j).
// logits[k,v] = sum_e mu_q_alpha[k,t,e] * word_emb[v,e]
// Branchless: rows >= KK / cols >= VV are clamped (consumers discard).
// E = 300 is an exact multiple of 4 -> no tail.
// =====================================================================
__device__ __forceinline__ void logits_tile(
    const float* __restrict__ alpha, const float* __restrict__ wemb,
    int t, int v0, int mt, int ng, int lane, v8f accs[4])
{
    const int half = lane >> 4;
    const int l16  = lane & 15;
    const int krow = mt * 16 + l16;
    const float* __restrict__ Ap =
        alpha + ((size_t)(krow < KK ? krow : KK - 1) * TT + t) * EE + 2 * half;
    const float* __restrict__ Bp0;
    const float* __restrict__ Bp1;
    const float* __restrict__ Bp2;
    const float* __restrict__ Bp3;
    {
        int vc0 = v0 + (ng * 4 + 0) * 16 + l16; if (vc0 >= VV) vc0 = VV - 1;
        int vc1 = v0 + (ng * 4 + 1) * 16 + l16; if (vc1 >= VV) vc1 = VV - 1;
        int vc2 = v0 + (ng * 4 + 2) * 16 + l16; if (vc2 >= VV) vc2 = VV - 1;
        int vc3 = v0 + (ng * 4 + 3) * 16 + l16; if (vc3 >= VV) vc3 = VV - 1;
        Bp0 = wemb + (size_t)vc0 * EE + 2 * half;
        Bp1 = wemb + (size_t)vc1 * EE + 2 * half;
        Bp2 = wemb + (size_t)vc2 * EE + 2 * half;
        Bp3 = wemb + (size_t)vc3 * EE + 2 * half;
    }
#pragma unroll 2
    for (int e = 0; e < EE; e += 4) {
        const v2f av = *(const v2f*)(Ap + e);
        const v2f b0 = *(const v2f*)(Bp0 + e);
        const v2f b1 = *(const v2f*)(Bp1 + e);
        const v2f b2 = *(const v2f*)(Bp2 + e);
        const v2f b3 = *(const v2f*)(Bp3 + e);
        accs[0] = __builtin_amdgcn_wmma_f32_16x16x4_f32(
            false, av, false, b0, (short)0, accs[0], false, false);
        accs[1] = __builtin_amdgcn_wmma_f32_16x16x4_f32(
            false, av, false, b1, (short)0, accs[1], false, false);
        accs[2] = __builtin_amdgcn_wmma_f32_16x16x4_f32(
            false, av, false, b2, (short)0, accs[2], false, false);
        accs[3] = __builtin_amdgcn_wmma_f32_16x16x4_f32(
            false, av, false, b3, (short)0, accs[3], false, false);
    }
}

// pass 1: partial row sums of exp(logits) -> zpart[t][chunk][ng][64]
__global__ __launch_bounds__(256)
void k_logits_zpart(const float* __restrict__ alpha,
                    const float* __restrict__ wemb,
                    float* __restrict__ zpart)
{
    const int t = blockIdx.y, chunk = blockIdx.x;
    const int v0 = chunk * VT;
    const int lane = threadIdx.x & 31;
    const int wave = threadIdx.x >> 5;
    const int mt = wave >> 1, ng = wave & 1;
    const int half = lane >> 4, l16 = lane & 15;

    v8f accs[4];
#pragma unroll
    for (int j = 0; j < 4; ++j) accs[j] = (v8f){0.f,0.f,0.f,0.f,0.f,0.f,0.f,0.f};
    logits_tile(alpha, wemb, t, v0, mt, ng, lane, accs);

    const size_t zbase = (((size_t)t * NCH + chunk) * 2 + ng) * 64;
#pragma unroll
    for (int i = 0; i < 8; ++i) {
        const int row = mt * 16 + 8 * half + i;
        float s = 0.f;
#pragma unroll
        for (int j = 0; j < 4; ++j) {
            const int vc = v0 + (ng * 4 + j) * 16 + l16;
            if (row < KK && vc < VV) s += expf(accs[j][i]);
        }
        for (int m = 1; m < 16; m <<= 1) s += __shfl_xor(s, m, 32);
        if (l16 == 0) zpart[zbase + row] = s;
    }
}

// reduce zpart -> Z[t*K+k]
__global__ __launch_bounds__(256)
void k_zreduce(const float* __restrict__ zpart, float* __restrict__ Z)
{
    const int idx = blockIdx.x * 256 + threadIdx.x;
    if (idx >= TT * KK) return;
    const int t = idx / KK, k = idx % KK;
    float s = 0.f;
    for (int ch = 0; ch < NCH; ++ch) {
        const size_t b = (((size_t)t * NCH + ch) * 2) * 64;
        s += zpart[b + k] + zpart[b + 64 + k];
    }
    Z[idx] = s;
}

// pass 2: recompute logits, normalize, accumulate sum_b,v bows*log(loglik+eps)
__global__ __launch_bounds__(256)
void k_nll(const float* __restrict__ alpha, const float* __restrict__ wemb,
           const float* __restrict__ Z, const float* __restrict__ theta,
           const float* __restrict__ bows, const int* __restrict__ times,
           float* __restrict__ nllpart)
{
    __shared__ float P[64][VT + 2];      // 33.3 KB
    __shared__ float th_s[BB * KK];      // 20 KB
    __shared__ float red[256];
    __shared__ int   ts[BB];

    const int t = blockIdx.y, chunk = blockIdx.x;
    const int v0 = chunk * VT;
    const int lane = threadIdx.x & 31;
    const int wave = threadIdx.x >> 5;
    const int mt = wave >> 1, ng = wave & 1;
    const int half = lane >> 4, l16 = lane & 15;

    for (int i = threadIdx.x; i < BB; i += 256) ts[i] = times[i];
    for (int i = threadIdx.x; i < BB * KK; i += 256) th_s[i] = theta[i];

    v8f accs[4];
#pragma unroll
    for (int j = 0; j < 4; ++j) accs[j] = (v8f){0.f,0.f,0.f,0.f,0.f,0.f,0.f,0.f};
    logits_tile(alpha, wemb, t, v0, mt, ng, lane, accs);

#pragma unroll
    for (int i = 0; i < 8; ++i) {
        const int row = mt * 16 + 8 * half + i;
        const float zi = (row < KK) ? 1.0f / Z[t * KK + row] : 0.f;
#pragma unroll
        for (int j = 0; j < 4; ++j) {
            const int cl = (ng * 4 + j) * 16 + l16;
            const int vc = v0 + cl;
            float p = 0.f;
            if (row < KK && vc < VV) p = expf(accs[j][i]) * zi;
            P[row][cl] = p;
        }
    }
    __syncthreads();

    float acc = 0.f;
    const int c = threadIdx.x;
    if (c < VT) {
        const int v = v0 + c;
        if (v < VV) {
            for (int b = 0; b < BB; ++b) {
                if (ts[b] != t) continue;
                float ll = 0.f;
#pragma unroll 5
                for (int k = 0; k < KK; ++k) ll += th_s[b * KK + k] * P[k][c];
                acc += bows[(size_t)b * VV + v] * logf(ll + EPS_C);
            }
        }
    }
    red[threadIdx.x] = acc;
    __syncthreads();
    for (int s = 128; s > 0; s >>= 1) {
        if (threadIdx.x < s) red[threadIdx.x] += red[threadIdx.x + s];
        __syncthreads();
    }
    if (threadIdx.x == 0) nllpart[(size_t)t * NCH + chunk] = red[0];
}

// =====================================================================
// Single-block sequential: 3-layer LSTM over T then eta chain + kl_eta
// =====================================================================
__global__ __launch_bounds__(800)
void k_lstm_eta(const float* __restrict__ out0,
                float* __restrict__ x0, float* __restrict__ x1,
                const float* __restrict__ Wih, const float* __restrict__ Whh,
                const float* __restrict__ bih, const float* __restrict__ bhh,
                const float* __restrict__ Wmu, const float* __restrict__ bmu,
                const float* __restrict__ Wls, const float* __restrict__ bls,
                float* __restrict__ etas, float* __restrict__ kl_eta_out)
{
    __shared__ float hs[EHH], cs[EHH], xs[EHH], gl[4 * EHH];
    __shared__ float mus[KK], lss[KK], kt[KK], eprev[KK];
    const int tid = threadIdx.x;

    for (int l = 0; l < LL; ++l) {
        const float* xin  = (l == 0) ? out0 : ((l == 1) ? x0 : x1);
        float*       xout = (l == 0) ? x0   : ((l == 1) ? x1 : x0);
        if (tid < EHH) { hs[tid] = 0.f; cs[tid] = 0.f; }
        __syncthreads();
        const float* Wi = Wih + (size_t)l * 4 * EHH * EHH;
        const float* Wh = Whh + (size_t)l * 4 * EHH * EHH;
        const float* bi = bih + l * 4 * EHH;
        const float* bh = bhh + l * 4 * EHH;
        for (int t = 0; t < TT; ++t) {
            if (tid < EHH) xs[tid] = xin[t * EHH + tid];
            __syncthreads();
            float acc = bi[tid] + bh[tid];
            const float* wi = Wi + (size_t)tid * EHH;
            const float* wh = Wh + (size_t)tid * EHH;
            for (int e = 0; e < EHH; ++e) acc += wi[e] * xs[e] + wh[e] * hs[e];
            gl[tid] = acc;
            __syncthreads();
            if (tid < EHH) {
                const float gi = gl[tid], gf = gl[EHH + tid];
                const float gg = gl[2 * EHH + tid], go = gl[3 * EHH + tid];
                float c = sigf(gf) * cs[tid] + sigf(gi) * tanhf(gg);
                cs[tid] = c;
                hs[tid] = sigf(go) * tanhf(c);
            }
            __syncthreads();
            if (tid < EHH) xout[t * EHH + tid] = hs[tid];
            __syncthreads();
        }
    }
    // eta chain; final LSTM output lives in x0 (l0:out0->x0, l1:x0->x1, l2:x1->x0)
    const float* fo = x0;
    float klacc = 0.f;
    const int gk = tid >> 4;   // 0..49 (800/16)
    const int j  = tid & 15;
    const float logd = logf(DELTA_C);
    for (int t = 0; t < TT; ++t) {
        if (tid < EHH) xs[tid] = fo[t * EHH + tid];
        __syncthreads();
        float am = 0.f, al = 0.f;
        for (int i = j; i < EHH + KK; i += 16) {
            const float v = (i < EHH) ? xs[i] : ((t == 0) ? 0.f : eprev[i - EHH]);
            am += Wmu[gk * (EHH + KK) + i] * v;
            al += Wls[gk * (EHH + KK) + i] * v;
        }
        for (int m = 1; m < 16; m <<= 1) { am += __shfl_xor(am, m, 32); al += __shfl_xor(al, m, 32); }
        if (j == 0) { mus[gk] = am + bmu[gk]; lss[gk] = al + bls[gk]; }
        __syncthreads();
        if (tid < KK) {
            const float mu = mus[tid], ls = lss[tid];
            etas[t * KK + tid] = mu;
            float term;
            if (t == 0) term = 0.5f * ((expf(ls) + mu * mu) / (1.0f + EPS_C) - 1.0f - ls);
            else {
                const float d = mu - eprev[tid];
                term = 0.5f * ((expf(ls) + d * d) / (DELTA_C + EPS_C) - 1.0f + logd - ls);
            }
            kt[tid] = term;
        }
        __syncthreads();
        if (tid == 0) { float s = 0.f; for (int k = 0; k < KK; ++k) s += kt[k]; klacc += s; }
        __syncthreads();
        if (tid < KK) eprev[tid] = mus[tid];
        __syncthreads();
    }
    if (tid == 0) kl_eta_out[0] = klacc;
}

// build theta input: inp[b] = [normalized_bows[b] (V), etas[times[b]] (K), 0 pad]
__global__ __launch_bounds__(256)
void k_fill_inp(const float* __restrict__ nb, const float* __restrict__ etas,
                const int* __restrict__ times, float* __restrict__ inp)
{
    const size_t idx = (size_t)blockIdx.x * 256 + threadIdx.x;
    if (idx >= (size_t)BB * KPAD) return;
    const int b = (int)(idx / KPAD);
    const int c = (int)(idx % KPAD);
    float v;
    if (c < VV)            v = nb[(size_t)b * VV + c];
    else if (c < VV + KK)  v = etas[times[b] * KK + (c - VV)];
    else                   v = 0.f;
    inp[idx] = v;
}

// per-b: mu_t, ls_t from h, softmax->theta, kl_theta partial
__global__ __launch_bounds__(512)
void k_theta(const float* __restrict__ h,
             const float* __restrict__ Wmu, const float* __restrict__ bmu,
             const float* __restrict__ Wls, const float* __restrict__ bls,
             const float* __restrict__ etas, const int* __restrict__ times,
             float* __restrict__ theta, float* __restrict__ klpart)
{
    __shared__ float hsh[THH], mus[64], lss[64], kt[64], sm[2];
    const int b = blockIdx.x;
    for (int i = threadIdx.x; i < THH; i += 512) hsh[i] = h[(size_t)b * THH + i];
    __syncthreads();
    const int gk = threadIdx.x >> 3, j = threadIdx.x & 7;
    if (gk < KK) {
        float am = 0.f, al = 0.f;
        for (int e = j; e < THH; e += 8) {
            const float hv = hsh[e];
            am += Wmu[gk * THH + e] * hv;
            al += Wls[gk * THH + e] * hv;
        }
        for (int m = 1; m < 8; m <<= 1) { am += __shfl_xor(am, m, 32); al += __shfl_xor(al, m, 32); }
        if (j == 0) { mus[gk] = am + bmu[gk]; lss[gk] = al + bls[gk]; }
    }
    __syncthreads();
    if (threadIdx.x == 0) {
        float mx = mus[0];
        for (int k = 1; k < KK; ++k) mx = fmaxf(mx, mus[k]);
        float s = 0.f;
        for (int k = 0; k < KK; ++k) s += expf(mus[k] - mx);
        sm[0] = mx; sm[1] = s;
    }
    __syncthreads();
    if (threadIdx.x < KK)
        theta[b * KK + threadIdx.x] = expf(mus[threadIdx.x] - sm[0]) / sm[1];
    const int tb = times[b];
    if (threadIdx.x < KK) {
        const int k = threadIdx.x;
        const float mu = mus[k], ls = lss[k], et = etas[tb * KK + k];
        kt[k] = 0.5f * ((expf(ls) + (mu - et) * (mu - et)) / (1.0f + EPS_C) - 1.0f - ls);
    }
    __syncthreads();
    if (threadIdx.x == 0) { float s = 0.f; for (int k = 0; k < KK; ++k) s += kt[k]; klpart[b] = s; }
}

// kl_alpha elementwise + block partials
__global__ __launch_bounds__(256)
void k_klalpha(const float* __restrict__ mu, const float* __restrict__ ls,
               float* __restrict__ part)
{
    __shared__ float red[256];
    const int idx = blockIdx.x * 256 + threadIdx.x;
    float v = 0.f;
    if (idx < KLA_NEL) {
        const int t = (idx / EE) % TT;
        const float m = mu[idx], l = ls[idx];
        if (t == 0) v = 0.5f * ((expf(l) + m * m) / (1.0f + EPS_C) - 1.0f - l);
        else {
            const float d = m - mu[idx - EE];
            v = 0.5f * ((expf(l) + d * d) / (DELTA_C + EPS_C) - 1.0f + logf(DELTA_C) - l);
        }
    }
    red[threadIdx.x] = v;
    __syncthreads();
    for (int s = 128; s > 0; s >>= 1) {
        if (threadIdx.x < s) red[threadIdx.x] += red[threadIdx.x + s];
        __syncthreads();
    }
    if (threadIdx.x == 0) part[blockIdx.x] = red[0];
}

// deterministic single-block sum
__global__ __launch_bounds__(256)
void k_reduce(const float* __restrict__ src, int n, float* __restrict__ dst)
{
    __shared__ float red[256];
    float a = 0.f;
    for (int i = threadIdx.x; i < n; i += 256) a += src[i];
    red[threadIdx.x] = a;
    __syncthreads();
    for (int s = 128; s > 0; s >>= 1) {
        if (threadIdx.x < s) red[threadIdx.x] += red[threadIdx.x + s];
        __syncthreads();
    }
    if (threadIdx.x == 0) dst[0] = red[0];
}

__global__ void k_finalize(const float* __restrict__ S,
                           const float* __restrict__ kleta,
                           float* __restrict__ out)
{
    if (threadIdx.x == 0 && blockIdx.x == 0) {
        const float nll = -S[0];
        const float kla = S[1];
        const float klt = S[2];
        const float kle = kleta[0];
        out[0] = nll + kla + kle + klt;  // nelbo
        out[1] = nll;
        out[2] = kla;
        out[3] = kle;
        out[4] = klt;
    }
}

// =====================================================================
extern "C" void kernel_launch(void* const* d_in, const int* in_sizes, int n_in,
                              void* d_out, int out_size, void* d_ws, size_t ws_size,
                              hipStream_t stream) {
    const float* bows        = (const float*)d_in[0];
    const float* nbows       = (const float*)d_in[1];
    const float* rnn_inp     = (const float*)d_in[2];
    const float* word_emb    = (const float*)d_in[3];
    const float* mu_q_alpha  = (const float*)d_in[4];
    const float* ls_q_alpha  = (const float*)d_in[5];
    const float* W_theta     = (const float*)d_in[6];
    const float* b_theta     = (const float*)d_in[7];
    const float* W_mu_theta  = (const float*)d_in[8];
    const float* b_mu_theta  = (const float*)d_in[9];
    const float* W_ls_theta  = (const float*)d_in[10];
    const float* b_ls_theta  = (const float*)d_in[11];
    const float* W_map       = (const float*)d_in[12];
    const float* b_map       = (const float*)d_in[13];
    const float* lstm_Wih    = (const float*)d_in[14];
    const float* lstm_Whh    = (const float*)d_in[15];
    const float* lstm_bih    = (const float*)d_in[16];
    const float* lstm_bhh    = (const float*)d_in[17];
    const float* W_mu_eta    = (const float*)d_in[18];
    const float* b_mu_eta    = (const float*)d_in[19];
    const float* W_ls_eta    = (const float*)d_in[20];
    const float* b_ls_eta    = (const float*)d_in[21];
    const int*   times       = (const int*)d_in[22];

    float* ws  = (float*)d_ws;   // needs ~17.4 MB
    float* out = (float*)d_out;

    // 1) out0 = rnn_inp @ W_map.T + b_map   [40,200]
    k_gemm_wmma<<<dim3((TT + 31) / 32, (EHH + 63) / 64), 256, 0, stream>>>(
        rnn_inp, VV, W_map, VV, b_map, ws + OFF_OUT0, EHH, TT, EHH, VV, 0);

    // 2) 3-layer LSTM + eta chain + kl_eta
    k_lstm_eta<<<1, 800, 0, stream>>>(
        ws + OFF_OUT0, ws + OFF_X0, ws + OFF_X1,
        lstm_Wih, lstm_Whh, lstm_bih, lstm_bhh,
        W_mu_eta, b_mu_eta, W_ls_eta, b_ls_eta,
        ws + OFF_ETAS, ws + OFF_KLETA);

    // 3) theta input  [100, 30052] (zero-padded to KPAD)
    {
        const size_t tot = (size_t)BB * KPAD;
        k_fill_inp<<<(unsigned)((tot + 255) / 256), 256, 0, stream>>>(
            nbows, ws + OFF_ETAS, times, ws + OFF_INP);
    }

    // 4) h = tanh(inp @ W_theta.T + b_theta)   [100, 800], KdReal = 30050
    k_gemm_wmma<<<dim3((BB + 31) / 32, (THH + 63) / 64), 256, 0, stream>>>(
        ws + OFF_INP, KPAD, W_theta, VV + KK, b_theta,
        ws + OFF_H, THH, BB, THH, VV + KK, 1);

    // 5) theta + kl_theta partials
    k_theta<<<BB, 512, 0, stream>>>(
        ws + OFF_H, W_mu_theta, b_mu_theta, W_ls_theta, b_ls_theta,
        ws + OFF_ETAS, times, ws + OFF_THETA, ws + OFF_KLTH);
    k_reduce<<<1, 256, 0, stream>>>(ws + OFF_KLTH, BB, ws + OFF_S + 2);

    // 6) softmax denominators Z[t,k] (pass 1 over logits)
    k_logits_zpart<<<dim3(NCH, TT), 256, 0, stream>>>(
        mu_q_alpha, word_emb, ws + OFF_ZPART);
    k_zreduce<<<(TT * KK + 255) / 256, 256, 0, stream>>>(
        ws + OFF_ZPART, ws + OFF_Z);

    // 7) NLL (pass 2 over logits)
    k_nll<<<dim3(NCH, TT), 256, 0, stream>>>(
        mu_q_alpha, word_emb, ws + OFF_Z, ws + OFF_THETA,
        bows, times, ws + OFF_NLLP);
    k_reduce<<<1, 256, 0, stream>>>(ws + OFF_NLLP, TT * NCH, ws + OFF_S + 0);

    // 8) kl_alpha
    k_klalpha<<<KLA_BLK, 256, 0, stream>>>(mu_q_alpha, ls_q_alpha, ws + OFF_KLAP);
    k_reduce<<<1, 256, 0, stream>>>(ws + OFF_KLAP, KLA_BLK, ws + OFF_S + 1);

    // 9) finalize
    k_finalize<<<1, 32, 0, stream>>>(ws + OFF_S, ws + OFF_KLETA, out);
}